// Residual_49091476193386
// MI455X (gfx1250) — compile-verified
//
#include <hip/hip_runtime.h>

typedef __attribute__((ext_vector_type(2))) float v2f;
typedef __attribute__((ext_vector_type(4))) float v4f;
typedef __attribute__((ext_vector_type(8))) float v8f;

#define DIM 512
#define DD  (DIM * DIM)

// ---------------------------------------------------------------------------
// xr = relu(x)
// ---------------------------------------------------------------------------
__global__ void relu_kernel(const float* __restrict__ x, float* __restrict__ y, int n) {
    int t = blockIdx.x * blockDim.x + threadIdx.x;
    if (t < n) y[t] = fmaxf(x[t], 0.0f);
}

// ---------------------------------------------------------------------------
// S[o,i] = sum_c W[c,o,i]   (streams the big weight tensor once, float4-wide)
// One thread owns 4 consecutive (o,i) flat positions. 65536 threads total.
// ---------------------------------------------------------------------------
__global__ void colsum_kernel(const float* __restrict__ W, float* __restrict__ S, int C) {
    int t = blockIdx.x * blockDim.x + threadIdx.x;        // 0..65535
    size_t base = (size_t)t * 4;
    v4f acc = {0.0f, 0.0f, 0.0f, 0.0f};
    for (int c = 0; c < C; ++c) {
        const float* p = W + (size_t)c * DD + base;
        if (c + 1 < C)
            __builtin_prefetch(W + (size_t)(c + 1) * DD + base, 0, 0); // global_prefetch_b8
        v4f w = *reinterpret_cast<const v4f*>(p);
        acc += w;
    }
    *reinterpret_cast<v4f*>(S + base) = acc;
}

// ---------------------------------------------------------------------------
// node[i] = argmax_o S[o,i], first index wins ties (strict > like jnp.argmax)
// Column-major walk is coalesced across the 512 i-threads.
// ---------------------------------------------------------------------------
__global__ void argmax_kernel(const float* __restrict__ S, int* __restrict__ node) {
    int i = blockIdx.x * blockDim.x + threadIdx.x;
    if (i >= DIM) return;
    float best = S[i];
    int bi = 0;
    for (int o = 1; o < DIM; ++o) {
        float v = S[(size_t)o * DIM + i];
        if (v > best) { best = v; bi = o; }
    }
    node[i] = bi;
}

// ---------------------------------------------------------------------------
// vals[c,i] = W[c, node[i], i] * xin[c,i]
// ---------------------------------------------------------------------------
__global__ void vals_kernel(const float* __restrict__ W, const float* __restrict__ xin,
                            const int* __restrict__ node, float* __restrict__ vals, int n) {
    int t = blockIdx.x * blockDim.x + threadIdx.x;
    if (t >= n) return;
    int c = t >> 9;          // / 512
    int i = t & (DIM - 1);   // % 512
    vals[t] = W[(size_t)c * DD + (size_t)node[i] * DIM + i] * xin[t];
}

// ---------------------------------------------------------------------------
// diff[c,n] = sum_{i : node[i]==n} vals[c,i]
// One block per channel c; node/vals row staged in LDS; deterministic
// ascending-i accumulation order (no float atomics).
// ---------------------------------------------------------------------------
__global__ void scatter_kernel(const int* __restrict__ node, const float* __restrict__ vals,
                               float* __restrict__ diff) {
    __shared__ int   ns[DIM];
    __shared__ float vs[DIM];
    int c = blockIdx.x;
    int t = threadIdx.x;               // 512 threads
    ns[t] = node[t];
    vs[t] = vals[(size_t)c * DIM + t];
    __syncthreads();
    float acc = 0.0f;
    for (int i = 0; i < DIM; ++i)
        if (ns[i] == t) acc += vs[i];  // LDS broadcast reads
    diff[(size_t)c * DIM + t] = acc;
}

// ---------------------------------------------------------------------------
// out = op( Wc (MxK) @ diff (Kx512) + bias )  [+ resid]
// One wave per 16x16 tile; V_WMMA_F32_16X16X4_F32, K stepped by 4.
// Per ISA layouts (lane = 16*hi + r, VGPR j): A/B element K = 2*hi + j,
// A row M = r, B col N = r; C/D: M = v + 8*hi, N = r.
// Grids are exact so EXEC is all-1s in every WMMA wave.
// ---------------------------------------------------------------------------
__global__ void gemm_wmma_kernel(const float* __restrict__ A,     // M x K  (Wc)
                                 const float* __restrict__ B,     // K x 512 (diff)
                                 const float* __restrict__ bias,  // M
                                 const float* __restrict__ resid, // M x 512 or nullptr
                                 float* __restrict__ out,         // M x 512
                                 int K, int do_relu) {
    int gtid = blockIdx.x * blockDim.x + threadIdx.x;
    int wave = gtid >> 5;
    int lane = gtid & 31;
    int tn = wave & 31;            // 512/16 = 32 column tiles
    int tm = wave >> 5;
    int m0 = tm * 16, n0 = tn * 16;
    int r  = lane & 15;
    int hi = lane >> 4;

    v8f c = {0.0f, 0.0f, 0.0f, 0.0f, 0.0f, 0.0f, 0.0f, 0.0f};
    const float* arow = A + (size_t)(m0 + r) * K;
    for (int k0 = 0; k0 < K; k0 += 4) {
        int ka = k0 + 2 * hi;
        v2f a = { arow[ka], arow[ka + 1] };
        v2f b = { B[(size_t)ka * DIM + n0 + r],
                  B[(size_t)(ka + 1) * DIM + n0 + r] };
        // 8 args: (neg_a, A, neg_b, B, c_mod, C, reuse_a, reuse_b)
        c = __builtin_amdgcn_wmma_f32_16x16x4_f32(false, a, false, b,
                                                  (short)0, c, false, false);
    }

#pragma unroll
    for (int v = 0; v < 8; ++v) {
        int m = m0 + v + 8 * hi;
        int n = n0 + r;
        float val = c[v] + bias[m];
        if (do_relu) val = fmaxf(val, 0.0f);
        if (resid)   val += resid[(size_t)m * DIM + n];
        out[(size_t)m * DIM + n] = val;
    }
}

// ---------------------------------------------------------------------------
extern "C" void kernel_launch(void* const* d_in, const int* in_sizes, int n_in,
                              void* d_out, int out_size, void* d_ws, size_t ws_size,
                              hipStream_t stream) {
    const float* x   = (const float*)d_in[0];   // (1,128,512)
    const float* W1  = (const float*)d_in[1];   // (128,512,512)
    const float* Wc1 = (const float*)d_in[2];   // (64,128)
    const float* bc1 = (const float*)d_in[3];   // (64,)
    const float* W2  = (const float*)d_in[4];   // (64,512,512)
    const float* Wc2 = (const float*)d_in[5];   // (128,64)
    const float* bc2 = (const float*)d_in[6];   // (128,)
    float* out = (float*)d_out;                 // (1,128,512)

    float* ws   = (float*)d_ws;
    float* xr   = ws;                           // 128*512
    float* S    = xr + 128 * DIM;               // 512*512
    int*   node = (int*)(S + DD);               // 512
    float* vals = (float*)(node + DIM);         // 128*512
    float* diff = vals + 128 * DIM;             // 128*512
    float* h1r  = diff + 128 * DIM;             // 64*512 (relu(h1))

    // xr = relu(x)
    relu_kernel<<<(128 * DIM) / 256, 256, 0, stream>>>(x, xr, 128 * DIM);

    // ---- stage 1: C=128 -> 64 ----
    colsum_kernel<<<DD / 4 / 256, 256, 0, stream>>>(W1, S, 128);
    argmax_kernel<<<2, 256, 0, stream>>>(S, node);
    vals_kernel<<<(128 * DIM) / 256, 256, 0, stream>>>(W1, xr, node, vals, 128 * DIM);
    scatter_kernel<<<128, DIM, 0, stream>>>(node, vals, diff);
    // h1r = relu(Wc1 @ diff + bc1): 4x32 tiles = 128 waves = 16 blocks
    gemm_wmma_kernel<<<16, 256, 0, stream>>>(Wc1, diff, bc1, nullptr, h1r, 128, 1);

    // ---- stage 2: C=64 -> 128 ----
    colsum_kernel<<<DD / 4 / 256, 256, 0, stream>>>(W2, S, 64);
    argmax_kernel<<<2, 256, 0, stream>>>(S, node);
    vals_kernel<<<(64 * DIM) / 256, 256, 0, stream>>>(W2, h1r, node, vals, 64 * DIM);
    scatter_kernel<<<64, DIM, 0, stream>>>(node, vals, diff);
    // out = Wc2 @ diff + bc2 + xr: 8x32 tiles = 256 waves = 32 blocks
    gemm_wmma_kernel<<<32, 256, 0, stream>>>(Wc2, diff, bc2, xr, out, 64, 0);
}